// GCN_3083786518791
// MI455X (gfx1250) — compile-verified
//
#include <hip/hip_runtime.h>
#include <hip/hip_bf16.h>

typedef float v2f __attribute__((ext_vector_type(2)));
typedef float v8f __attribute__((ext_vector_type(8)));

#define N_NODES 100000

// ---------------------------------------------------------------------------
// fp32 WMMA GEMM: C[M,N] = A[M,K] * B[K,N].  One wave (32 lanes) per 16x16
// C tile, K stepped by 4 using V_WMMA_F32_16X16X4_F32.
// A-layout per ISA: lanes 0-15 row M=lane (K=k0,k0+1), lanes 16-31 same rows
// (K=k0+2,k0+3).  B-layout mirrored: lanes 0-15 col N=lane rows k0/k0+1,
// lanes 16-31 rows k0+2/k0+3.  C: VGPR v holds M = 8*half + v, N = lane&15.
// M,N,K all multiples of 16/4 here (100000 % 16 == 0) -> no guards.
// ---------------------------------------------------------------------------
__global__ __launch_bounds__(32)
void gemm_wmma_f32(const float* __restrict__ A, const float* __restrict__ B,
                   float* __restrict__ C, int N, int K) {
  const int tn = blockIdx.x;            // N tile
  const int tm = blockIdx.y;            // M tile
  const int lane = threadIdx.x & 31;
  const int half = lane >> 4;           // 0: K=k0..k0+1, 1: K=k0+2..k0+3
  const int l    = lane & 15;

  const float* aPtr = A + (size_t)((tm << 4) + l) * K + 2 * half;   // row-major A
  const float* bPtr = B + (size_t)(2 * half) * N + (tn << 4) + l;   // row-major B

  v8f acc = {};
  #pragma unroll 8
  for (int k0 = 0; k0 < K; k0 += 4) {
    v2f a = *(const v2f*)(aPtr + k0);               // A[row, k0+2h .. k0+2h+1]
    v2f b;
    b.x = bPtr[(size_t)k0 * N];                     // B[k0+2h,   col]
    b.y = bPtr[(size_t)k0 * N + N];                 // B[k0+2h+1, col]
    acc = __builtin_amdgcn_wmma_f32_16x16x4_f32(
        /*neg_a=*/false, a, /*neg_b=*/false, b,
        /*c_mod=*/(short)0, acc, /*reuse_a=*/false, /*reuse_b=*/false);
  }

  float* cPtr = C + (size_t)((tm << 4) + (half << 3)) * N + (tn << 4) + l;
  #pragma unroll
  for (int v = 0; v < 8; ++v) cPtr[(size_t)v * N] = acc[v];
}

// ---------------------------------------------------------------------------
// Degree computation (in-degree incl. self-loop) and dinv = rsqrt(deg)
// ---------------------------------------------------------------------------
__global__ void init_deg(float* deg, int n) {
  int i = blockIdx.x * blockDim.x + threadIdx.x;
  if (i < n) deg[i] = 1.0f;             // self-loop
}

__global__ void deg_edges(const long long* __restrict__ dst, float* deg, int nE) {
  int e = blockIdx.x * blockDim.x + threadIdx.x;
  if (e < nE) unsafeAtomicAdd(&deg[dst[e]], 1.0f);  // hw GLOBAL_ATOMIC_ADD_F32
}

__global__ void deg_to_dinv(float* deg, int n) {
  int i = blockIdx.x * blockDim.x + threadIdx.x;
  if (i < n) deg[i] = rsqrtf(deg[i]);   // deg >= 1 always
}

// ---------------------------------------------------------------------------
// out[n,f] = h[n,f] * dinv[n]^2 (+ bias[f]) : self-loop term, float4 wide
// ---------------------------------------------------------------------------
__global__ void self_loop_init(const float* __restrict__ h,
                               const float* __restrict__ dinv,
                               const float* __restrict__ bias,   // may be null
                               float* __restrict__ out, int n, int F) {
  int i4 = blockIdx.x * blockDim.x + threadIdx.x;  // float4 index
  long long total = (long long)n * F / 4;
  if (i4 >= total) return;
  int node = (int)(((long long)i4 * 4) / F);
  int f0   = (int)(((long long)i4 * 4) % F);
  float w = dinv[node] * dinv[node];
  float4 hv = ((const float4*)h)[i4];
  float4 o;
  o.x = hv.x * w; o.y = hv.y * w; o.z = hv.z * w; o.w = hv.w * w;
  if (bias) {
    float4 bv = *(const float4*)(bias + f0);
    o.x += bv.x; o.y += bv.y; o.z += bv.z; o.w += bv.w;
  }
  ((float4*)out)[i4] = o;
}

// ---------------------------------------------------------------------------
// Edge scatter: one wave per edge; lane owns F/32 contiguous floats.
// out[dst,f] += h[src,f] * dinv[src]*dinv[dst] via hw fp32 atomics
// (unsafeAtomicAdd -> GLOBAL_ATOMIC_ADD_F32, no CAS loop).
// ---------------------------------------------------------------------------
template <int F>
__global__ void scatter_edges(const float* __restrict__ h,
                              float* __restrict__ out,
                              const long long* __restrict__ src,
                              const long long* __restrict__ dst,
                              const float* __restrict__ dinv, int nE) {
  int e = blockIdx.x * (blockDim.x >> 5) + (threadIdx.x >> 5);
  if (e >= nE) return;
  int lane = threadIdx.x & 31;
  long long s = src[e], d = dst[e];
  float w = dinv[s] * dinv[d];
  constexpr int VPL = F / 32;  // 4 (F=128) or 2 (F=64)
  const float* hp = h + (size_t)s * F + lane * VPL;
  float* op = out + (size_t)d * F + lane * VPL;
  if constexpr (VPL == 4) {
    float4 hv = *(const float4*)hp;     // coalesced 128b gather
    unsafeAtomicAdd(op + 0, hv.x * w);
    unsafeAtomicAdd(op + 1, hv.y * w);
    unsafeAtomicAdd(op + 2, hv.z * w);
    unsafeAtomicAdd(op + 3, hv.w * w);
  } else {
    float2 hv = *(const float2*)hp;
    unsafeAtomicAdd(op + 0, hv.x * w);
    unsafeAtomicAdd(op + 1, hv.y * w);
  }
}

// ---------------------------------------------------------------------------
// x = relu(x + b[f])  over [n, F], float4 wide
// ---------------------------------------------------------------------------
__global__ void bias_relu(float* __restrict__ x, const float* __restrict__ b,
                          int n, int F) {
  int i4 = blockIdx.x * blockDim.x + threadIdx.x;
  long long total = (long long)n * F / 4;
  if (i4 >= total) return;
  int f0 = (int)(((long long)i4 * 4) % F);
  float4 v  = ((float4*)x)[i4];
  float4 bv = *(const float4*)(b + f0);
  v.x = fmaxf(v.x + bv.x, 0.0f);
  v.y = fmaxf(v.y + bv.y, 0.0f);
  v.z = fmaxf(v.z + bv.z, 0.0f);
  v.w = fmaxf(v.w + bv.w, 0.0f);
  ((float4*)x)[i4] = v;
}

// ---------------------------------------------------------------------------
extern "C" void kernel_launch(void* const* d_in, const int* in_sizes, int n_in,
                              void* d_out, int out_size, void* d_ws, size_t ws_size,
                              hipStream_t stream) {
  const float*     x   = (const float*)d_in[0];
  const long long* ei  = (const long long*)d_in[1];   // int64 [2, E]
  const float*     W1  = (const float*)d_in[2];
  const float*     b1  = (const float*)d_in[3];
  const float*     W2  = (const float*)d_in[4];
  const float*     b2  = (const float*)d_in[5];
  float*           out = (float*)d_out;

  const int n  = N_NODES;
  const int nE = in_sizes[1] / 2;       // 600000
  const long long* src = ei;
  const long long* dstp = ei + nE;

  // workspace: deg/dinv @0 (400KB), A @1MiB (51.2MB), B @51MiB (51.2MB)
  char* w = (char*)d_ws;
  float* dinv = (float*)w;
  float* hA = (float*)(w + (1ull << 20));
  float* hB = (float*)(w + (1ull << 20) + 52428800ull);

  const int TPB = 256;

  // ---- degrees -> dinv ----
  init_deg<<<(n + TPB - 1) / TPB, TPB, 0, stream>>>(dinv, n);
  deg_edges<<<(nE + TPB - 1) / TPB, TPB, 0, stream>>>(dstp, dinv, nE);
  deg_to_dinv<<<(n + TPB - 1) / TPB, TPB, 0, stream>>>(dinv, n);

  // ---- layer 1: h1 = x @ W1  (100000x128 @ 128x128) ----
  {
    dim3 grid(128 / 16, n / 16);        // (8, 6250) tiles, 1 wave each
    gemm_wmma_f32<<<grid, 32, 0, stream>>>(x, W1, hA, 128, 128);
  }
  {
    long long q4 = (long long)n * 128 / 4;
    self_loop_init<<<(unsigned)((q4 + TPB - 1) / TPB), TPB, 0, stream>>>(
        hA, dinv, nullptr, hB, n, 128);
    scatter_edges<128><<<(nE + 7) / 8, 256, 0, stream>>>(hA, hB, src, dstp, dinv, nE);
    bias_relu<<<(unsigned)((q4 + TPB - 1) / TPB), TPB, 0, stream>>>(hB, b1, n, 128);
  }

  // ---- layer 2: h2 = relu(h1) @ W2  (100000x128 @ 128x64) ----
  {
    dim3 grid(64 / 16, n / 16);         // (4, 6250)
    gemm_wmma_f32<<<grid, 32, 0, stream>>>(hB, W2, hA, 64, 128);
  }
  {
    long long q4 = (long long)n * 64 / 4;
    self_loop_init<<<(unsigned)((q4 + TPB - 1) / TPB), TPB, 0, stream>>>(
        hA, dinv, b2, out, n, 64);      // fully overwrites d_out (bias fused)
    scatter_edges<64><<<(nE + 7) / 8, 256, 0, stream>>>(hA, out, src, dstp, dinv, nE);
  }
}